// Model_79748952752235
// MI455X (gfx1250) — compile-verified
//
#include <hip/hip_runtime.h>
#include <hip/hip_bf16.h>

typedef __attribute__((ext_vector_type(2))) float v2f;
typedef __attribute__((ext_vector_type(8))) float v8f;

#define NPART 2048
#define DIM   32
#define NSQ   (NPART * NPART)           // 4194304
#define MED_RANK ((NSQ - 1) / 2)        // 2097151
#define LOGN_F 7.62461899f              // ln(2048), > 1 so max(ln N, 1) = ln N

// ---------------------------------------------------------------- sq[i] = |x_i|^2
__global__ void sq_kernel(const float* __restrict__ P, float* __restrict__ sq) {
    int i = blockIdx.x * blockDim.x + threadIdx.x;   // 8*256 = 2048
    const float* row = P + i * DIM;
    float s = 0.f;
#pragma unroll
    for (int d = 0; d < DIM; ++d) s += row[d] * row[d];
    sq[i] = s;
}

// ---------------------------------------------------------------- radix-select state init
__global__ void init_state(unsigned* __restrict__ state, unsigned* __restrict__ hist) {
    hist[threadIdx.x] = 0u;
    if (threadIdx.x == 0) { state[0] = 0u; state[1] = (unsigned)MED_RANK; }
}

// ---------------------------------------------------------------- dist^2 via f32 WMMA gram
// one 16x16 tile per wave; 8 waves per block; 2048 blocks -> 16384 tiles
__global__ void dist_kernel(const float* __restrict__ P, const float* __restrict__ sq,
                            float* __restrict__ dist) {
    const int tid  = threadIdx.x;
    const int w    = tid >> 5;
    const int lane = tid & 31;
    const int m    = lane & 15;            // row-in-tile for A, col-in-tile for B
    const int half = lane >> 4;            // 0: K lanes {0,1}, 1: K lanes {2,3}
    const int tile = blockIdx.x * 8 + w;
    const int ti   = tile >> 7;            // 128 tiles per row
    const int tj   = tile & 127;

    const int ai = ti * 16 + m;            // global row for A loads
    const int bi = tj * 16 + m;            // global row for B loads (B = P^T tile)
    const int koff = half * 2;

    v8f acc = {};
#pragma unroll
    for (int kk = 0; kk < DIM; kk += 4) {
        v2f a = *(const v2f*)(P + ai * DIM + kk + koff);
        v2f b = *(const v2f*)(P + bi * DIM + kk + koff);
        acc = __builtin_amdgcn_wmma_f32_16x16x4_f32(false, a, false, b,
                                                    (short)0, acc, false, false);
    }

    const int j = tj * 16 + m;
    const float sqj = sq[j];
#pragma unroll
    for (int v = 0; v < 8; ++v) {
        const int i = ti * 16 + v + half * 8;
        float d = sq[i] + sqj - 2.0f * acc[v];
        d = d > 0.0f ? d : 0.0f;           // clamp; guarantees +0 bit pattern
        dist[i * NPART + j] = d;
    }
}

// ---------------------------------------------------------------- radix-select histogram
__global__ void hist_kernel(const float* __restrict__ dist, const unsigned* __restrict__ state,
                            unsigned* __restrict__ hist, int shift) {
    __shared__ unsigned lh[256];
    const int tid = threadIdx.x;
    lh[tid] = 0u;
    __syncthreads();
    const unsigned pref = state[0];
    const unsigned mask = (shift == 24) ? 0u : (0xFFFFFFFFu << (shift + 8));
#pragma unroll
    for (int t = 0; t < 4; ++t) {
        const int idx = blockIdx.x * 1024 + t * 256 + tid;   // 4096 blocks cover 4194304
        const unsigned bits = __float_as_uint(dist[idx]);
        if ((bits & mask) == (pref & mask))
            atomicAdd(&lh[(bits >> shift) & 255u], 1u);
    }
    __syncthreads();
    const unsigned c = lh[tid];
    if (c) atomicAdd(&hist[tid], c);
}

// ---------------------------------------------------------------- pick bucket, update state
__global__ void select_kernel(unsigned* __restrict__ state, unsigned* __restrict__ hist,
                              float* __restrict__ hstate, int shift) {
    __shared__ unsigned sh[256];
    const int tid = threadIdx.x;
    sh[tid] = hist[tid];
    __syncthreads();
    hist[tid] = 0u;                         // ready for next round
    if (tid == 0) {
        unsigned k = state[1];
        unsigned cum = 0u;
        unsigned pref = state[0];
        for (int b = 0; b < 256; ++b) {
            const unsigned c = sh[b];
            if (cum + c > k) {
                pref |= ((unsigned)b) << shift;
                state[0] = pref;
                state[1] = k - cum;
                if (shift == 0) {
                    float med = __uint_as_float(pref);
                    med = fmaxf(med, 1e-5f);
                    const float h = med / LOGN_F;
                    hstate[0] = h;
                    hstate[1] = 0.5f / h;   // 1/(2h)
                    hstate[2] = 1.0f / h;
                }
                break;
            }
            cum += c;
        }
    }
}

// ---------------------------------------------------------------- fused K = exp(), ksum, K@P, phi
// one 16-row slab of output per block; 8 waves split the i (contraction) range
__global__ void phi_kernel(const float* __restrict__ P, const float* __restrict__ dist,
                           const float* __restrict__ hstate, float* __restrict__ out) {
    __shared__ float accs[8][512];          // per-wave 16x32 partial KP slabs
    __shared__ float ks[8][32];             // per-wave per-lane K partial sums
    __shared__ float ksum_s[16];

    const int tid  = threadIdx.x;
    const int w    = tid >> 5;
    const int lane = tid & 31;
    const int m    = lane & 15;
    const int half = lane >> 4;
    const int jt   = blockIdx.x;            // 128 row-tiles
    const int j    = jt * 16 + m;           // A row (output row) this lane feeds
    const float inv2h = hstate[1];

    v8f acc0 = {};
    v8f acc1 = {};
    float ssum = 0.f;
    const int ibeg = w * 256;
#pragma unroll 4
    for (int ib = 0; ib < 256; ib += 4) {
        const int i0 = ibeg + ib + half * 2;
        // A tile of K computed in registers: K[j, i0], K[j, i0+1]
        const float d0 = dist[j * NPART + i0];
        const float d1 = dist[j * NPART + i0 + 1];
        const float k0 = __expf(-d0 * inv2h);
        const float k1 = __expf(-d1 * inv2h);
        v2f a; a.x = k0; a.y = k1;
        ssum += k0 + k1;
        // B tiles: P rows i0,i0+1, dim cols m and m+16
        v2f b0, b1;
        b0.x = P[i0 * DIM + m];        b0.y = P[(i0 + 1) * DIM + m];
        b1.x = P[i0 * DIM + m + 16];   b1.y = P[(i0 + 1) * DIM + m + 16];
        acc0 = __builtin_amdgcn_wmma_f32_16x16x4_f32(false, a, false, b0,
                                                     (short)0, acc0, false, false);
        acc1 = __builtin_amdgcn_wmma_f32_16x16x4_f32(false, a, false, b1,
                                                     (short)0, acc1, false, false);
    }

#pragma unroll
    for (int v = 0; v < 8; ++v) {
        const int M = v + half * 8;
        accs[w][M * 32 + m]      = acc0[v];
        accs[w][M * 32 + m + 16] = acc1[v];
    }
    ks[w][lane] = ssum;
    __syncthreads();

    // reduce the 8 partial slabs: each thread owns elements tid and tid+256
    float r0 = 0.f, r1 = 0.f;
#pragma unroll
    for (int ww = 0; ww < 8; ++ww) { r0 += accs[ww][tid]; r1 += accs[ww][tid + 256]; }
    accs[0][tid] = r0;
    accs[0][tid + 256] = r1;
    if (tid < 16) {
        float s = 0.f;
#pragma unroll
        for (int ww = 0; ww < 8; ++ww) s += ks[ww][tid] + ks[ww][tid + 16];
        ksum_s[tid] = s;                    // ksum for row jt*16 + tid
    }
    __syncthreads();

    const float invh = hstate[2];
    const float invN = 1.0f / (float)NPART;
#pragma unroll
    for (int e = tid; e < 512; e += 256) {
        const int jj = e >> 5, d = e & 31;
        const int gj = jt * 16 + jj;
        const float kp = accs[0][e];                    // (K @ P)[gj, d]
        const float p  = P[gj * DIM + d];
        // phi = (K@score + grad_K_sum)/N ; K@score = -KP
        const float phi = (-kp - (p * ksum_s[jj] - kp) * invh) * invN;
        out[gj * DIM + d] = phi;
    }
}

extern "C" void kernel_launch(void* const* d_in, const int* in_sizes, int n_in,
                              void* d_out, int out_size, void* d_ws, size_t ws_size,
                              hipStream_t stream) {
    const float* P = (const float*)d_in[0];
    float* out = (float*)d_out;

    float*    dist   = (float*)d_ws;                 // 4194304 floats (16 MB)
    float*    sq     = dist + NSQ;                   // 2048 floats
    unsigned* state  = (unsigned*)(sq + NPART);      // pref, k
    unsigned* hist   = state + 2;                    // 256 bins
    float*    hstate = (float*)(hist + 256);         // h, 1/(2h), 1/h

    sq_kernel  <<<NPART / 256, 256, 0, stream>>>(P, sq);
    init_state <<<1, 256, 0, stream>>>(state, hist);
    dist_kernel<<<2048, 256, 0, stream>>>(P, sq, dist);

    for (int shift = 24; shift >= 0; shift -= 8) {
        hist_kernel  <<<4096, 256, 0, stream>>>(dist, state, hist, shift);
        select_kernel<<<1, 256, 0, stream>>>(state, hist, hstate, shift);
    }

    phi_kernel<<<128, 256, 0, stream>>>(P, dist, hstate, out);
}